// EdgeNetwork_87127706566742
// MI455X (gfx1250) — compile-verified
//
#include <hip/hip_runtime.h>
#include <hip/hip_bf16.h>
#include <math.h>

// Problem constants (match the reference).
#define B_ 4
#define N_ 2048
#define E_ 8192
#define F_ 16
#define H_ 8
#define BE_ (B_ * E_)          // 32768 edges total
#define NCHUNK 64              // rows of the one-hot scanned per thread
#define NCHUNKS (N_ / NCHUNK)  // 32

typedef __attribute__((ext_vector_type(2))) float v2f;
typedef __attribute__((ext_vector_type(8))) float v8f;

// ---------------------------------------------------------------------------
// Zero the index-accumulator scratch (harness poisons d_ws; we re-zero every
// call so the launch is deterministic).
// ---------------------------------------------------------------------------
__global__ __launch_bounds__(256)
void zero_f32_kernel(float* __restrict__ p, int n) {
    int i = blockIdx.x * 256 + threadIdx.x;
    if (i < n) p[i] = 0.0f;
}

// ---------------------------------------------------------------------------
// Streaming one-hot scan: acc[b,e] = sum_n n * R[b,n,e]  (== the index).
// Each thread owns 4 consecutive edges (float4, 16B aligned, fully coalesced
// across the wave) and a 64-row chunk of n. Only the chunk containing the 1
// produces a nonzero partial, so the atomicAdd is taken ~once per column.
// This pass IS the whole cost of the problem: 2 x 268MB at HBM rate.
// ---------------------------------------------------------------------------
__global__ __launch_bounds__(256)
void scan_onehot_kernel(const float* __restrict__ R, float* __restrict__ acc) {
    const int tid   = blockIdx.x * 256 + threadIdx.x;   // 0 .. (BE/4)*NCHUNKS-1
    const int c4    = tid & (BE_ / 4 - 1);              // edge-quad id
    const int chunk = tid >> 13;                        // n-chunk id (BE/4 = 8192 = 2^13)
    const int b     = c4 >> 11;                         // E/4 = 2048 = 2^11
    const int e4    = (c4 & 2047) << 2;                 // first of 4 edges
    const int n0    = chunk * NCHUNK;

    const float* base = R + (size_t)b * N_ * E_ + e4;
    float4 p = make_float4(0.f, 0.f, 0.f, 0.f);
#pragma unroll 4
    for (int i = 0; i < NCHUNK; ++i) {
        const int n = n0 + i;
        const float4 v = *(const float4*)(base + (size_t)n * E_);
        const float fn = (float)n;
        p.x = fmaf(fn, v.x, p.x);
        p.y = fmaf(fn, v.y, p.y);
        p.z = fmaf(fn, v.z, p.z);
        p.w = fmaf(fn, v.w, p.w);
    }
    float* o = acc + (size_t)b * E_ + e4;
    if (p.x != 0.0f) atomicAdd(o + 0, p.x);
    if (p.y != 0.0f) atomicAdd(o + 1, p.y);
    if (p.z != 0.0f) atomicAdd(o + 2, p.z);
    if (p.w != 0.0f) atomicAdd(o + 3, p.w);
}

// ---------------------------------------------------------------------------
// Per-edge MLP via WMMA. One wave (32 lanes) handles a tile of 16 edges:
//   D(16x16) = A(16x32 Bcat) x B(32x16 = W1^T zero-padded), via 8 chained
//   V_WMMA_F32_16X16X4_F32 steps (K=4 each).
// A layout (32-bit, 16x4): lane L supplies A[L%16][4s+2*(L/16)] and +1.
// B layout (symmetric):    lane L supplies B[4s+2*(L/16)][L%16] and +1,
//   where B[k][n] = (n<8) ? W1[n][k] : 0  (masked by multiply, no branches).
// D layout: vgpr r, lane L -> M = r + 8*(L/16), N = L%16.
// Epilogue: tanh(+b1[N]) * W2[N], xor-shuffle reduce over the 16-lane half,
// lanes 0/16 write the 16 sigmoids.
// ---------------------------------------------------------------------------
__global__ __launch_bounds__(32)
void mlp_wmma_kernel(const float* __restrict__ X,
                     const float* __restrict__ srcAcc,
                     const float* __restrict__ dstAcc,
                     const float* __restrict__ W1,
                     const float* __restrict__ b1,
                     const float* __restrict__ W2,
                     const float* __restrict__ b2,
                     float* __restrict__ out) {
    const int lane = threadIdx.x;          // 0..31
    const int m    = lane & 15;            // A row / B-D column
    const int hi   = lane >> 4;            // half select (K/M offset)
    const int tile = blockIdx.x;           // BE/16 tiles
    const int b    = tile / (E_ / 16);
    const int e0   = (tile % (E_ / 16)) * 16;
    const int edge = e0 + m;

    const int si = (int)(srcAcc[(size_t)b * E_ + edge] + 0.5f);
    const int di = (int)(dstAcc[(size_t)b * E_ + edge] + 0.5f);
    const float* xs = X + ((size_t)b * N_ + si) * F_;   // Bcat[:,0:16]  (bo, from Ro)
    const float* xd = X + ((size_t)b * N_ + di) * F_;   // Bcat[:,16:32] (bi, from Ri)

    const float wmask = (m < H_) ? 1.0f : 0.0f;         // zero-pad columns 8..15
    const int   hrow  = m & (H_ - 1);
    const float* w1r  = W1 + hrow * (2 * F_);

    v8f c = {0.f, 0.f, 0.f, 0.f, 0.f, 0.f, 0.f, 0.f};
#pragma unroll
    for (int s = 0; s < 8; ++s) {
        const int ka = 4 * s + 2 * hi;                  // absolute k of A.x
        v2f a, bb;
        if (s < 4) { a.x = xs[ka];      a.y = xs[ka + 1]; }
        else       { a.x = xd[ka - 16]; a.y = xd[ka - 15]; }
        bb.x = w1r[ka]     * wmask;
        bb.y = w1r[ka + 1] * wmask;
        c = __builtin_amdgcn_wmma_f32_16x16x4_f32(
                /*neg_a=*/false, a, /*neg_b=*/false, bb,
                /*c_mod=*/(short)0, c, /*reuse_a=*/false, /*reuse_b=*/false);
    }

    const float bias  = b1[hrow] * wmask;
    const float w2v   = W2[hrow] * wmask;
    const float bias2 = b2[0];

    float sig[8];
#pragma unroll
    for (int r = 0; r < 8; ++r) {
        float t = tanhf(c[r] + bias) * w2v;             // exact 0 for N>=8
#pragma unroll
        for (int mk = 1; mk < 16; mk <<= 1)
            t += __shfl_xor(t, mk, 32);                 // reduce over h within half
        sig[r] = t;
    }
    if (m == 0) {                                       // lanes 0 and 16 emit
        float* op = out + (size_t)b * E_ + e0 + 8 * hi;
#pragma unroll
        for (int r = 0; r < 8; ++r)
            op[r] = 1.0f / (1.0f + expf(-(sig[r] + bias2)));
    }
}

// ---------------------------------------------------------------------------
extern "C" void kernel_launch(void* const* d_in, const int* in_sizes, int n_in,
                              void* d_out, int out_size, void* d_ws, size_t ws_size,
                              hipStream_t stream) {
    (void)in_sizes; (void)n_in; (void)out_size; (void)ws_size;
    const float* X  = (const float*)d_in[0];
    const float* Ri = (const float*)d_in[1];   // dst one-hot
    const float* Ro = (const float*)d_in[2];   // src one-hot
    const float* W1 = (const float*)d_in[3];
    const float* b1 = (const float*)d_in[4];
    const float* W2 = (const float*)d_in[5];
    const float* b2 = (const float*)d_in[6];
    float* out = (float*)d_out;

    float* srcAcc = (float*)d_ws;              // [BE] recovered src indices
    float* dstAcc = srcAcc + BE_;              // [BE] recovered dst indices

    zero_f32_kernel<<<(2 * BE_ + 255) / 256, 256, 0, stream>>>(srcAcc, 2 * BE_);

    const int scan_blocks = (BE_ / 4) * NCHUNKS / 256;  // 1024
    scan_onehot_kernel<<<scan_blocks, 256, 0, stream>>>(Ro, srcAcc);
    scan_onehot_kernel<<<scan_blocks, 256, 0, stream>>>(Ri, dstAcc);

    mlp_wmma_kernel<<<BE_ / 16, 32, 0, stream>>>(X, srcAcc, dstAcc,
                                                 W1, b1, W2, b2, out);
}